// NKLM_66348654788985
// MI455X (gfx1250) — compile-verified
//
#include <hip/hip_runtime.h>
#include <math.h>

// NKLM forward on MI455X (gfx1250).
// GEMMs: OUT[M x 16] = W[M x K] @ X[K x 16], batch B=16 = WMMA N dim, fp32 WMMA 16x16x4.
// Activations are stored batch-major [16][K] so both A and B fragments are contiguous
// float2 (b64) loads and the C/D epilogue is two b128 stores per lane.
// Weights (~130MB) are L2-resident (192MB L2), so steps 1..95 stream from L2.

#define BB 16
#define TT 96
#define VV 32000
#define DW 512
#define DF 512
#define HH 1024
#define LL 100
#define FN 64
#define EPSF 1e-8f

typedef __attribute__((ext_vector_type(2))) float v2f;
typedef __attribute__((ext_vector_type(4))) float v4f;
typedef __attribute__((ext_vector_type(8))) float v8f;

__device__ __forceinline__ v8f wmma_f32_k4(v2f a, v2f b, v8f c) {
    return __builtin_amdgcn_wmma_f32_16x16x4_f32(false, a, false, b, (short)0, c, false, false);
}

// ---------------- WMMA GEMM: OUT[16 x M] = (accum?OUT:0) + (W[M x K] @ X^T)^T + bias, opt relu
// X batch-major [16][K]; OUT batch-major [16][M].
__global__ void nklm_gemm16(const float* __restrict__ W, const float* __restrict__ X,
                            const float* __restrict__ bias, float* __restrict__ OUT,
                            int M, int K, int relu, int accum) {
    int wave = (int)((blockIdx.x * blockDim.x + threadIdx.x) >> 5);
    int lane = (int)(threadIdx.x & 31);
    int m0 = wave * 16;
    if (m0 >= M) return;                        // wave-uniform: EXEC stays all-1 for WMMA
    int half = lane >> 4, l15 = lane & 15;
    int row = m0 + l15;
    if (row >= M) row = M - 1;                  // clamp: rows >= M produce garbage in unstored
    const v2f* wr = (const v2f*)(W + (size_t)row * (size_t)K);   // D rows -- no masking needed
    const v2f* xb = (const v2f*)(X + (size_t)l15 * (size_t)K);
    v8f acc0 = {}, acc1 = {};
    int k = 0;
#pragma unroll 2
    for (; k + 8 <= K; k += 8) {                // two independent acc chains for WMMA ILP
        int i0 = (k >> 1) + half, i1 = i0 + 2;
        v2f a0 = wr[i0], b0 = xb[i0];
        v2f a1 = wr[i1], b1 = xb[i1];
        acc0 = wmma_f32_k4(a0, b0, acc0);
        acc1 = wmma_f32_k4(a1, b1, acc1);
    }
    for (; k < K; k += 4) {
        int i0 = (k >> 1) + half;
        acc0 = wmma_f32_k4(wr[i0], xb[i0], acc0);
    }
    int r0 = m0 + 8 * half;                     // lane's 8 consecutive output rows
    float* o = OUT + (size_t)l15 * (size_t)M + r0;
    float res[8];
#pragma unroll
    for (int v = 0; v < 8; ++v) res[v] = acc0[v] + acc1[v];
    if (m0 + 16 <= M) {                         // full tile: vector epilogue (wave-uniform branch)
        if (bias) {
            v4f blo = *(const v4f*)(bias + r0), bhi = *(const v4f*)(bias + r0 + 4);
#pragma unroll
            for (int v = 0; v < 4; ++v) { res[v] += blo[v]; res[v + 4] += bhi[v]; }
        }
        if (accum) {
            v4f olo = *(const v4f*)(o), ohi = *(const v4f*)(o + 4);
#pragma unroll
            for (int v = 0; v < 4; ++v) { res[v] += olo[v]; res[v + 4] += ohi[v]; }
        }
        v4f slo, shi;
#pragma unroll
        for (int v = 0; v < 4; ++v) {
            slo[v] = relu ? fmaxf(res[v], 0.0f) : res[v];
            shi[v] = relu ? fmaxf(res[v + 4], 0.0f) : res[v + 4];
        }
        *(v4f*)(o) = slo;
        *(v4f*)(o + 4) = shi;
    } else {                                    // edge tile (M=1, M=100 tails): guarded scalar
#pragma unroll
        for (int v = 0; v < 8; ++v) {
            int r = r0 + v;
            if (r < M) {
                float val = res[v];
                if (bias)  val += bias[r];
                if (accum) val += o[v];
                if (relu)  val = fmaxf(val, 0.0f);
                o[v] = val;
            }
        }
    }
}

// ---------------- vocab: per-wave 16x16 tile of emb @ k_voca, fused exp-sum -> sumexp[16]
__global__ void nklm_vocab(const float* __restrict__ emb, const float* __restrict__ kv,
                           float* __restrict__ sumexp) {
    int wave = (int)((blockIdx.x * blockDim.x + threadIdx.x) >> 5);
    int lane = (int)(threadIdx.x & 31);
    int m0 = wave * 16;
    if (m0 >= VV) return;
    int half = lane >> 4, l15 = lane & 15;
    const v2f* wr = (const v2f*)(emb + (size_t)(m0 + l15) * DW);
    const v2f* xb = (const v2f*)(kv + (size_t)l15 * DW);
    v8f acc0 = {}, acc1 = {};
#pragma unroll 4
    for (int k = 0; k < DW; k += 8) {
        int i0 = (k >> 1) + half, i1 = i0 + 2;
        v2f a0 = wr[i0], b0 = xb[i0];
        v2f a1 = wr[i1], b1 = xb[i1];
        acc0 = wmma_f32_k4(a0, b0, acc0);
        acc1 = wmma_f32_k4(a1, b1, acc1);
    }
    float s = 0.0f;
#pragma unroll
    for (int v = 0; v < 8; ++v) s += __expf(acc0[v] + acc1[v]);   // scores O(1): no max-shift
    s += __shfl_down(s, 16, 32);                 // fold hi-half rows (same batch column)
    if (lane < 16) atomicAdd(&sumexp[l15], s);
}

// ---------------- LSTM pointwise: gates[16][4096] (+b_hh) -> h,c; h mirrored into oe/oa
__global__ void nklm_lstm_pw(const float* __restrict__ gates, const float* __restrict__ b_hh,
                             float* __restrict__ h, float* __restrict__ c,
                             float* __restrict__ oe, float* __restrict__ oa) {
    int idx = (int)(blockIdx.x * blockDim.x + threadIdx.x);
    if (idx >= HH * BB) return;
    int b = idx >> 10, j = idx & (HH - 1);
    const float* g = gates + (size_t)b * 4 * HH;
    float ig = g[j]          + b_hh[j];
    float fg = g[HH + j]     + b_hh[HH + j];
    float gg = g[2 * HH + j] + b_hh[2 * HH + j];
    float og = g[3 * HH + j] + b_hh[3 * HH + j];
    float si = 1.0f / (1.0f + __expf(-ig));
    float sf = 1.0f / (1.0f + __expf(-fg));
    float so = 1.0f / (1.0f + __expf(-og));
    size_t sidx = (size_t)b * HH + j;
    float cn = sf * c[sidx] + si * tanhf(gg);
    float hn = so * tanhf(cn);
    c[sidx] = cn; h[sidx] = hn;
    oe[(size_t)b * (HH + DF) + j] = hn;
    oa[(size_t)b * (HH + DF) + j] = hn;
}

// ---------------- attention: softmax over 64 facts, mask+renorm, argmax, gather a_vec
__global__ void nklm_attn(const float* __restrict__ fe, const int* __restrict__ true_a,
                          const int* __restrict__ nf_arr, const float* __restrict__ k_fact,
                          float* __restrict__ oa, float* __restrict__ x_cat,
                          float* __restrict__ out_loss, int t) {
    int b = (int)blockIdx.x;      // 16 blocks
    int f = (int)threadIdx.x;     // 64 threads
    __shared__ float sc[FN];
    __shared__ float mbuf[FN];
    __shared__ int aidx_s;
    const float* fb = fe + (size_t)(b * FN + f) * DF;
    const float* kf = k_fact + (size_t)b * DF;
    float dot = 0.0f;
    for (int d = 0; d < DF; ++d) dot += fb[d] * kf[d];
    sc[f] = dot;
    __syncthreads();
    if (f == 0) {
        float mx = sc[0];
        for (int i = 1; i < FN; ++i) mx = fmaxf(mx, sc[i]);
        float se = 0.0f;
        for (int i = 0; i < FN; ++i) se += __expf(sc[i] - mx);
        int nf = nf_arr[b];
        float ssum = 0.0f, best = -1.0f; int bi = 0;
        for (int i = 0; i < FN; ++i) {
            float mask = (i == FN - 1 || i < nf) ? 1.0f : 0.0f;
            float m = mask * __expf(sc[i] - mx) / se + EPSF;
            mbuf[i] = m; ssum += m;
            if (m > best) { best = m; bi = i; }   // first-max tie-break like jnp.argmax
        }
        aidx_s = bi;
        int at = true_a[b * TT + t];
        atomicAdd(out_loss, -(__logf(mbuf[at]) - __logf(ssum)));
    }
    __syncthreads();
    const float* av = fe + (size_t)(b * FN + aidx_s) * DF;
    for (int d = f; d < DF; d += FN) {
        float v = av[d];
        oa[(size_t)b * (HH + DF) + HH + d] = v;   // oa = [h ; a_vec]
        x_cat[(size_t)b * (DF + DW + LL) + d] = v; // a_prev for next step
    }
}

// ---------------- finalize: vocab/pos/z losses + next-step wv/wo; one warp per batch item
__global__ void nklm_final(const float* __restrict__ emb, const float* __restrict__ pos_mask,
                           const int* __restrict__ true_w, const int* __restrict__ true_a,
                           const int* __restrict__ true_z, const float* __restrict__ k_voca,
                           const float* __restrict__ zlogit, const float* __restrict__ pos_logits,
                           float* __restrict__ sumexp, float* __restrict__ x_cat,
                           float* __restrict__ out_loss, int t) {
    int b = (int)(threadIdx.x >> 5);
    int lane = (int)(threadIdx.x & 31);
    int wt = true_w[b * TT + t], at = true_a[b * TT + t], zt = true_z[b * TT + t];
    const float* kv = k_voca + (size_t)b * DW;
    const float* pl = pos_logits + (size_t)b * LL;
    // s[wt] = emb[wt] . k_voca[b,:]
    float dot = 0.0f;
    for (int d = lane; d < DW; d += 32) dot += emb[(size_t)wt * DW + d] * kv[d];
    // pos softmax sums
    const float* pm = pos_mask + (size_t)(b * FN + at) * LL;
    float se = 0.0f, sm = 0.0f;
    for (int l = lane; l < LL; l += 32) {
        float e = __expf(pl[l]);
        se += e; sm += pm[l] * e;
    }
    for (int off = 16; off > 0; off >>= 1) {
        dot += __shfl_down(dot, off, 32);
        se  += __shfl_down(se,  off, 32);
        sm  += __shfl_down(sm,  off, 32);
    }
    if (lane == 0) {
        float lse = __logf(sumexp[b]);
        sumexp[b] = 0.0f;                                  // re-arm for next step's vocab pass
        float loss;
        if (zt == 0) {
            loss = -(dot - lse);                           // -log_w[wt]
        } else {
            float mpsum = sm / se + (float)LL * EPSF;
            float mwt = pm[wt] * __expf(pl[wt]) / se + EPSF;
            loss = -(__logf(mwt) - __logf(mpsum));         // -log_pos[wt]
        }
        float x = zlogit[b];
        float z = 1.0f / (1.0f + __expf(-x));
        float zf = (float)zt;
        loss += -(zf * __logf(z) + (1.0f - zf) * __logf(1.0f - z));
        atomicAdd(out_loss, loss);
    }
    // next-step LSTM inputs: wv at x_cat[b][512..1023], wo at x_cat[b][1024..1123]
    float* xc = x_cat + (size_t)b * (DF + DW + LL);
    for (int d = lane; d < DW; d += 32)
        xc[DF + d] = (zt == 0) ? emb[(size_t)wt * DW + d] : 0.0f;
    for (int l = lane; l < LL; l += 32)
        xc[DF + DW + l] = (zt != 0 && l == wt) ? 1.0f : 0.0f;
}

// ---------------- e_k (once): mean of valid fact embeddings -> oe[b][1024..1535]
__global__ void nklm_ek(const float* __restrict__ fe, const int* __restrict__ nf_arr,
                        float* __restrict__ oe) {
    int idx = (int)(blockIdx.x * blockDim.x + threadIdx.x);
    if (idx >= DF * BB) return;
    int b = idx >> 9, d = idx & (DF - 1);
    int nf = nf_arr[b];
    float s = 0.0f;
    for (int f = 0; f < nf; ++f) s += fe[(size_t)(b * FN + f) * DF + d];
    oe[(size_t)b * (HH + DF) + HH + d] = s / (float)nf;
}

// ---------------- init: zero state/loss, wv0 = emb[SOS]
__global__ void nklm_init(const float* __restrict__ emb, float* __restrict__ x_cat,
                          float* __restrict__ h, float* __restrict__ c,
                          float* __restrict__ sumexp, float* __restrict__ out) {
    const int NX = (DF + DW + LL) * BB;   // 17984
    const int NH = HH * BB;               // 16384
    int idx = (int)(blockIdx.x * blockDim.x + threadIdx.x);
    if (idx < NX) {
        int r = idx % (DF + DW + LL);
        float v = 0.0f;
        if (r >= DF && r < DF + DW) v = emb[r - DF];   // emb[SOS=0][d]
        x_cat[idx] = v;
    } else if (idx < NX + NH) {
        h[idx - NX] = 0.0f;
    } else if (idx < NX + 2 * NH) {
        c[idx - NX - NH] = 0.0f;
    } else if (idx < NX + 2 * NH + BB) {
        sumexp[idx - NX - 2 * NH] = 0.0f;
    } else if (idx == NX + 2 * NH + BB) {
        out[0] = 0.0f;
    }
}

extern "C" void kernel_launch(void* const* d_in, const int* in_sizes, int n_in,
                              void* d_out, int out_size, void* d_ws, size_t ws_size,
                              hipStream_t stream) {
    const float* fe   = (const float*)d_in[0];
    const float* pmsk = (const float*)d_in[1];
    const float* emb  = (const float*)d_in[2];
    const float* W_ih = (const float*)d_in[3];
    const float* W_hh = (const float*)d_in[4];
    const float* b_ih = (const float*)d_in[5];
    const float* b_hh = (const float*)d_in[6];
    const float* Wf1  = (const float*)d_in[7];  const float* bf1 = (const float*)d_in[8];
    const float* Wf2  = (const float*)d_in[9];  const float* bf2 = (const float*)d_in[10];
    const float* Wc1  = (const float*)d_in[11]; const float* bc1 = (const float*)d_in[12];
    const float* Wc2  = (const float*)d_in[13]; const float* bc2 = (const float*)d_in[14];
    const float* Wv1  = (const float*)d_in[15]; const float* bv1 = (const float*)d_in[16];
    const float* Wv2  = (const float*)d_in[17]; const float* bv2 = (const float*)d_in[18];
    const float* Wp1  = (const float*)d_in[19]; const float* bp1 = (const float*)d_in[20];
    const float* Wp2  = (const float*)d_in[21]; const float* bp2 = (const float*)d_in[22];
    const float* WP   = (const float*)d_in[23]; const float* bP  = (const float*)d_in[24];
    const int* true_w = (const int*)d_in[25];
    const int* true_a = (const int*)d_in[26];
    const int* true_z = (const int*)d_in[27];
    const int* nf     = (const int*)d_in[28];
    float* out = (float*)d_out;

    // workspace carve-up (floats), activations batch-major [16][features]
    float* ws = (float*)d_ws;
    size_t off = 0;
    float* x_cat  = ws + off; off += (size_t)BB * (DF + DW + LL);
    float* h_buf  = ws + off; off += (size_t)BB * HH;
    float* c_buf  = ws + off; off += (size_t)BB * HH;
    float* gates  = ws + off; off += (size_t)BB * 4 * HH;
    float* oe     = ws + off; off += (size_t)BB * (HH + DF);   // [h ; e_k]
    float* t1     = ws + off; off += (size_t)BB * HH;
    float* k_fact = ws + off; off += (size_t)BB * DF;
    float* oa     = ws + off; off += (size_t)BB * (HH + DF);   // [h ; a_vec]
    float* t2     = ws + off; off += (size_t)BB * HH;
    float* zlog   = ws + off; off += BB;
    float* t3     = ws + off; off += (size_t)BB * HH;
    float* k_voca = ws + off; off += (size_t)BB * DW;
    float* t4     = ws + off; off += (size_t)BB * HH;
    float* k_pos  = ws + off; off += (size_t)BB * LL;
    float* plog   = ws + off; off += (size_t)BB * LL;
    float* sumexp = ws + off; off += BB;

    auto gemm = [&](const float* W, const float* X, const float* bias, float* OUT,
                    int M, int K, int relu, int accum) {
        int tiles = (M + 15) / 16;
        int blocks = (tiles + 7) / 8;     // 8 waves (256 threads) per block
        nklm_gemm16<<<blocks, 256, 0, stream>>>(W, X, bias, OUT, M, K, relu, accum);
    };

    {
        int n = (DF + DW + LL) * BB + 2 * HH * BB + BB + 1;
        nklm_init<<<(n + 255) / 256, 256, 0, stream>>>(emb, x_cat, h_buf, c_buf, sumexp, out);
        nklm_ek<<<(DF * BB + 255) / 256, 256, 0, stream>>>(fe, nf, oe);
    }

    for (int t = 0; t < TT; ++t) {
        // LSTM gates = W_ih @ x + b_ih  +=  W_hh @ h        (b_hh added pointwise)
        gemm(W_ih, x_cat, b_ih, gates, 4 * HH, DF + DW + LL, 0, 0);
        gemm(W_hh, h_buf, nullptr, gates, 4 * HH, HH, 0, 1);
        nklm_lstm_pw<<<(HH * BB + 255) / 256, 256, 0, stream>>>(gates, b_hh, h_buf, c_buf, oe, oa);
        // fact-key MLP + attention
        gemm(Wf1, oe, bf1, t1, HH, HH + DF, 1, 0);
        gemm(Wf2, t1, bf2, k_fact, DF, HH, 0, 0);
        nklm_attn<<<BB, FN, 0, stream>>>(fe, true_a, nf, k_fact, oa, x_cat, out, t);
        // copy / vocab / pos heads off oa
        gemm(Wc1, oa, bc1, t2, HH, HH + DF, 1, 0);
        gemm(Wc2, t2, bc2, zlog, 1, HH, 0, 0);
        gemm(Wv1, oa, bv1, t3, HH, HH + DF, 1, 0);
        gemm(Wv2, t3, bv2, k_voca, DW, HH, 0, 0);
        gemm(Wp1, oa, bp1, t4, HH, HH + DF, 1, 0);
        gemm(Wp2, t4, bp2, k_pos, LL, HH, 0, 0);
        gemm(WP, k_pos, bP, plog, LL, LL, 0, 0);
        // 32000x16 vocab tile pass with fused logsumexp (2000 waves)
        nklm_vocab<<<(VV / 16 + 7) / 8, 256, 0, stream>>>(emb, k_voca, sumexp);
        nklm_final<<<1, BB * 32, 0, stream>>>(emb, pmsk, true_w, true_a, true_z, k_voca,
                                              zlog, plog, sumexp, x_cat, out, t);
    }
    (void)in_sizes; (void)n_in; (void)out_size; (void)ws_size;
}